// QuantumStyleEncoder_39324720562420
// MI455X (gfx1250) — compile-verified
//
#include <hip/hip_runtime.h>
#include <hip/hip_bf16.h>

#define WIRES 6
#define QLAYERS 2
#define DIM 64
#define INPUT_DIM 128
#define LN_EPS 1e-5f
#define ROWS_PER_WAVE 16
#define WAVES_PER_BLOCK 8
#define TSTRIDE 68   // LDS row stride in floats (even -> 8B aligned float2)

typedef float v2f __attribute__((ext_vector_type(2)));
typedef float v8f __attribute__((ext_vector_type(8)));

// ---------------------------------------------------------------------------
// Kernel 1: fuse the 12 Rot gates + 12 CNOTs into one 64x64 complex operator.
// Thread j pushes basis vector e_j through the circuit: state = C[:, j].
// Output layout: packed WMMA-B tiles for 16x16x4 f32:
//   float index = ((kk*4 + ntile)*32 + lane)*2 + par
//   holds B[k = 4*kk + 2*(lane/16) + par][n = 16*ntile + (lane%16)]
//   where B[k][n] = C^T[k][n] = C[n][k].
// ---------------------------------------------------------------------------
__global__ __launch_bounds__(64) void qenc_build_operator(
    const float* __restrict__ qw,   // (QLAYERS, WIRES, 3)
    float* __restrict__ Pre, float* __restrict__ Pim)
{
    const int j = threadIdx.x;      // basis index / column of C / row of B
    float sr[DIM], si[DIM];
#pragma unroll
    for (int i = 0; i < DIM; ++i) { sr[i] = (i == j) ? 1.0f : 0.0f; si[i] = 0.0f; }

#pragma unroll
    for (int l = 0; l < QLAYERS; ++l) {
#pragma unroll
        for (int w = 0; w < WIRES; ++w) {
            const float phi = qw[(l * WIRES + w) * 3 + 0];
            const float th  = qw[(l * WIRES + w) * 3 + 1];
            const float om  = qw[(l * WIRES + w) * 3 + 2];
            float c, s, cA, sA, cB, sB;
            sincosf(0.5f * th, &s, &c);
            sincosf(0.5f * (phi + om), &sA, &cA);
            sincosf(0.5f * (phi - om), &sB, &cB);
            // Rot = RZ(om) RY(th) RZ(phi):
            // u00 = e^{-iA} c, u01 = -e^{+iB} s, u10 = e^{-iB} s, u11 = e^{+iA} c
            const float u00r =  cA * c, u00i = -sA * c;
            const float u01r = -cB * s, u01i = -sB * s;
            const float u10r =  cB * s, u10i = -sB * s;
            const float u11r =  cA * c, u11i =  sA * c;
            const int b = 5 - w;            // wire w <-> flat-index bit (5-w)
#pragma unroll
            for (int i = 0; i < DIM; ++i) {
                if ((i >> b) & 1) continue;
                const int i1 = i | (1 << b);
                const float ar = sr[i],  ai = si[i];
                const float br = sr[i1], bi = si[i1];
                sr[i]  = u00r*ar - u00i*ai + u01r*br - u01i*bi;
                si[i]  = u00r*ai + u00i*ar + u01r*bi + u01i*br;
                sr[i1] = u10r*ar - u10i*ai + u11r*br - u11i*bi;
                si[i1] = u10r*ai + u10i*ar + u11r*bi + u11i*br;
            }
        }
        const int rr = l % (WIRES - 1) + 1;   // CNOT ring range
#pragma unroll
        for (int w = 0; w < WIRES; ++w) {
            const int cb = 5 - w;
            const int tb = 5 - ((w + rr) % WIRES);
#pragma unroll
            for (int i = 0; i < DIM; ++i) {
                if (((i >> cb) & 1) && !((i >> tb) & 1)) {
                    const int i1 = i | (1 << tb);
                    float t;
                    t = sr[i]; sr[i] = sr[i1]; sr[i1] = t;
                    t = si[i]; si[i] = si[i1]; si[i1] = t;
                }
            }
        }
    }

    // scatter row j of B=C^T into packed WMMA-B layout
    const int kk   = j >> 2;
    const int half = (j >> 1) & 1;
    const int par  = j & 1;
#pragma unroll
    for (int i = 0; i < DIM; ++i) {
        const int lane = half * 16 + (i & 15);
        const int idx  = ((kk * 4 + (i >> 4)) * 32 + lane) * 2 + par;
        Pre[idx] = sr[i];
        Pim[idx] = si[i];
    }
}

// ---------------------------------------------------------------------------
// Kernel 2: fused pre-net + state build + WMMA f32 GEMM vs C^T + |.|^2 + signs.
// One wave of 32 lanes handles a 16-row batch tile.
// ---------------------------------------------------------------------------
__global__ __launch_bounds__(256) void qenc_main(
    const float* __restrict__ x,      // (B,128)
    const float* __restrict__ Wmat,   // (6,128)
    const float* __restrict__ bvec,   // (6)
    const float* __restrict__ gvec,   // (6)
    const float* __restrict__ bevec,  // (6)
    const float* __restrict__ Pre,    // packed B-tiles, 64*64
    const float* __restrict__ Pim,
    float* __restrict__ out)          // (B,6)
{
    __shared__ float sW[WIRES * INPUT_DIM];
    __shared__ float sTile[WAVES_PER_BLOCK][ROWS_PER_WAVE * TSTRIDE]; // amps, then probs

    const int tid  = threadIdx.x;
    const int wave = tid >> 5;
    const int lane = tid & 31;
    const int r    = lane & 15;    // row within tile
    const int h    = lane >> 4;    // K-half / lane-half

    for (int i = tid; i < WIRES * INPUT_DIM; i += 256) sW[i] = Wmat[i];
    __syncthreads();

    const int tileRow = (blockIdx.x * WAVES_PER_BLOCK + wave) * ROWS_PER_WAVE;

    // ---- pre-net: z = x @ W^T + b (each lane: one row, half of K) ----
    float acc[WIRES] = {0.f, 0.f, 0.f, 0.f, 0.f, 0.f};
    const float4* xrow = (const float4*)(x + (size_t)(tileRow + r) * INPUT_DIM + h * 64);
#pragma unroll
    for (int k4 = 0; k4 < 16; ++k4) {
        const float4 xv = xrow[k4];
#pragma unroll
        for (int c = 0; c < WIRES; ++c) {
            const float* wr = sW + c * INPUT_DIM + h * 64 + k4 * 4;
            acc[c] += xv.x * wr[0] + xv.y * wr[1] + xv.z * wr[2] + xv.w * wr[3];
        }
    }
    float z[WIRES];
    float mu = 0.f;
#pragma unroll
    for (int c = 0; c < WIRES; ++c) {
        acc[c] += __shfl_xor(acc[c], 16, 32);   // combine K-halves
        z[c] = acc[c] + bvec[c];
        mu += z[c];
    }
    mu *= (1.0f / WIRES);
    float var = 0.f;
#pragma unroll
    for (int c = 0; c < WIRES; ++c) { const float d = z[c] - mu; var += d * d; }
    var *= (1.0f / WIRES);
    const float inv = rsqrtf(var + LN_EPS);

    float cw[WIRES], sw[WIRES];
#pragma unroll
    for (int c = 0; c < WIRES; ++c) {
        const float y = tanhf(gvec[c] * (z[c] - mu) * inv + bevec[c]);
        sincosf(0.5f * y, &sw[c], &cw[c]);
    }

    // ---- amplitude product tree: lane writes 32 amps of its row ----
    float* amp = &sTile[wave][0];
    {
        const float p0 = h ? sw[0] : cw[0];     // wire0 <-> bit5 == h
        float a1[2], a2[4], a3[8], a4[16], a5[32];
        a1[0] = p0 * cw[1]; a1[1] = p0 * sw[1];
#pragma unroll
        for (int u = 0; u < 2; ++u)  { a2[2*u] = a1[u]*cw[2]; a2[2*u+1] = a1[u]*sw[2]; }
#pragma unroll
        for (int u = 0; u < 4; ++u)  { a3[2*u] = a2[u]*cw[3]; a3[2*u+1] = a2[u]*sw[3]; }
#pragma unroll
        for (int u = 0; u < 8; ++u)  { a4[2*u] = a3[u]*cw[4]; a4[2*u+1] = a3[u]*sw[4]; }
#pragma unroll
        for (int u = 0; u < 16; ++u) { a5[2*u] = a4[u]*cw[5]; a5[2*u+1] = a4[u]*sw[5]; }
#pragma unroll
        for (int t = 0; t < 32; ++t)
            amp[r * TSTRIDE + h * 32 + t] = a5[t];
    }
    __syncthreads();   // cross-lane LDS handoff (amps -> A fragments)

    // ---- GEMM: Y = S(16x64) @ C^T(64x64) on the matrix pipe, re & im ----
    v8f accRe[4], accIm[4];
    const v8f vzero = {0.f,0.f,0.f,0.f,0.f,0.f,0.f,0.f};
#pragma unroll
    for (int nt = 0; nt < 4; ++nt) { accRe[nt] = vzero; accIm[nt] = vzero; }

    const float2* PreV = (const float2*)Pre;
    const float2* PimV = (const float2*)Pim;
#pragma unroll
    for (int kk = 0; kk < 16; ++kk) {
        // A fragment: lane l -> S[m=r][k=4kk+2h+{0,1}]
        const float2 av = *(const float2*)(amp + r * TSTRIDE + 4 * kk + 2 * h);
        v2f A; A[0] = av.x; A[1] = av.y;
#pragma unroll
        for (int nt = 0; nt < 4; ++nt) {
            const float2 br = PreV[(kk * 4 + nt) * 32 + lane];
            const float2 bi = PimV[(kk * 4 + nt) * 32 + lane];
            v2f Br; Br[0] = br.x; Br[1] = br.y;
            v2f Bi; Bi[0] = bi.x; Bi[1] = bi.y;
            accRe[nt] = __builtin_amdgcn_wmma_f32_16x16x4_f32(
                false, A, false, Br, (short)0, accRe[nt], false, false);
            accIm[nt] = __builtin_amdgcn_wmma_f32_16x16x4_f32(
                false, A, false, Bi, (short)0, accIm[nt], false, false);
        }
    }

    // ---- probs = re^2 + im^2 from C-layout accumulators ----
    // C/D layout: VGPR v, lane: M = v + 8*h, N = nt*16 + r
    float* pT = &sTile[wave][0];   // reuse (same-wave LDS ops are in-order)
#pragma unroll
    for (int nt = 0; nt < 4; ++nt)
#pragma unroll
        for (int v = 0; v < 8; ++v) {
            const float re = accRe[nt][v];
            const float im = accIm[nt][v];
            pT[(v + 8 * h) * TSTRIDE + nt * 16 + r] = re * re + im * im;
        }
    __syncthreads();   // cross-lane LDS handoff (probs -> reduction)

    // ---- out[row][w] = sum_n probs[row][n] * (1 - 2*bit_{5-w}(n)); 32*3 = 96 vals
#pragma unroll
    for (int u = 0; u < 3; ++u) {
        const int idx = lane * 3 + u;
        const int row = idx / WIRES;
        const int w   = idx % WIRES;
        const int tb  = 5 - w;
        float ssum = 0.f;
#pragma unroll
        for (int n = 0; n < DIM; ++n) {
            const float pv = pT[row * TSTRIDE + n];
            ssum += ((n >> tb) & 1) ? -pv : pv;
        }
        out[(size_t)(tileRow + row) * WIRES + w] = ssum;
    }
}

// ---------------------------------------------------------------------------
extern "C" void kernel_launch(void* const* d_in, const int* in_sizes, int n_in,
                              void* d_out, int out_size, void* d_ws, size_t ws_size,
                              hipStream_t stream) {
    const float* x    = (const float*)d_in[0];   // inputs (B,128)
    const float* Wm   = (const float*)d_in[1];   // W (6,128)
    const float* bv   = (const float*)d_in[2];   // b (6)
    const float* gv   = (const float*)d_in[3];   // ln_gamma (6)
    const float* bev  = (const float*)d_in[4];   // ln_beta (6)
    const float* qw   = (const float*)d_in[5];   // q_weights (2,6,3)

    float* Pre = (float*)d_ws;                   // 64*64 f32
    float* Pim = Pre + DIM * DIM;                // 64*64 f32

    const int batch  = in_sizes[0] / INPUT_DIM;
    const int blocks = batch / (ROWS_PER_WAVE * WAVES_PER_BLOCK);

    qenc_build_operator<<<1, 64, 0, stream>>>(qw, Pre, Pim);
    qenc_main<<<blocks, WAVES_PER_BLOCK * 32, 0, stream>>>(
        x, Wm, bv, gv, bev, Pre, Pim, (float*)d_out);
}